// DeepLipschitzLinearResNet_46660524704118
// MI455X (gfx1250) — compile-verified
//
#include <hip/hip_runtime.h>
#include <cstddef>

// ---------------------------------------------------------------------------
// DeepLipschitzLinearResNet for MI455X (gfx1250), fp32 end-to-end.
// All dense matmuls run on V_WMMA_F32_16X16X4_F32 (the fp32 matrix pipe).
// LDS tiles use a pair-interleaved layout so each WMMA A/B fragment is one
// aligned ds_load_b64 straight into an even VGPR pair (no register shuffles).
// Cholesky / triangular-inverse are tiny (~5% of FLOPs) VALU kernels.
// ---------------------------------------------------------------------------

typedef float v2f __attribute__((ext_vector_type(2)));
typedef float v4f __attribute__((ext_vector_type(4)));
typedef float v8f __attribute__((ext_vector_type(8)));

#define BM 128
#define BN 128
#define BK 32
// pair-row stride in floats: 2*BM + 32  (mod 64 banks == 32 so the two lane
// groups' rows occupy disjoint bank halves -> conflict-free b64 reads)
#define AST (2 * BM + 32)
#define BST (2 * BN + 32)

enum { EPI_STORE = 0, EPI_IDENT = 1, EPI_ACCUM = 2, EPI_BIAS = 3, EPI_BIAS_RELU = 4 };

// ---------------------------------------------------------------------------
// WMMA fp32 GEMM: C[M,N] = alpha * op(A) @ op(B)  (+ epilogue)
//  - 256 threads = 8 waves (wave32), wave grid 2(M) x 4(N)
//  - each wave computes a 64x32 region = 4x2 WMMA 16x16 tiles
//  - LDS pair-interleaved: elem (k,c) at [(k>>1)*STRIDE + c*2 + (k&1)]
// grid = (N/BN, M/BM); all dims multiples of the tiles (1024 / 4096).
// ---------------------------------------------------------------------------
template <bool TA, bool TB>
__global__ __launch_bounds__(256) void gemm_f32_wmma(
    const float* __restrict__ A, const float* __restrict__ B,
    const float* __restrict__ bias, float* __restrict__ C,
    int K, int lda, int ldb, int ldc, float alpha, int epi)
{
    __shared__ float As[(BK / 2) * AST];
    __shared__ float Bs[(BK / 2) * BST];

    const int tid   = threadIdx.x;
    const int lane  = tid & 31;
    const int wave  = tid >> 5;
    const int waveM = wave >> 2;        // 0..1
    const int waveN = wave & 3;         // 0..3
    const int l15   = lane & 15;
    const int g     = (lane >> 4) & 1;  // ISA half-wave split
    const int m0    = blockIdx.y * BM;
    const int n0    = blockIdx.x * BN;

    const v8f zero8 = {0.f, 0.f, 0.f, 0.f, 0.f, 0.f, 0.f, 0.f};
    v8f acc[4][2];
#pragma unroll
    for (int i = 0; i < 4; ++i)
#pragma unroll
        for (int j = 0; j < 2; ++j) acc[i][j] = zero8;

    for (int k0 = 0; k0 < K; k0 += BK) {
        // ---- stage A tile (float4 global loads, paired LDS layout) ----
        if (TA) {
            // memory A[k][m]; vectorize over m
#pragma unroll
            for (int it = 0; it < (BK * BM / 4) / 256; ++it) {
                int t = tid + it * 256;
                int k = t / (BM / 4), mq = t % (BM / 4);
                v4f v = *(const v4f*)(A + (size_t)(k0 + k) * lda + (m0 + 4 * mq));
                float* p = &As[(k >> 1) * AST + (4 * mq) * 2 + (k & 1)];
                p[0] = v.x; p[2] = v.y; p[4] = v.z; p[6] = v.w;
            }
        } else {
            // memory A[m][k]; vectorize over k
#pragma unroll
            for (int it = 0; it < (BM * BK / 4) / 256; ++it) {
                int t = tid + it * 256;
                int m = t / (BK / 4), q = t % (BK / 4);
                v4f v = *(const v4f*)(A + (size_t)(m0 + m) * lda + (k0 + 4 * q));
                *(v2f*)&As[(2 * q) * AST + m * 2]     = (v2f){v.x, v.y};
                *(v2f*)&As[(2 * q + 1) * AST + m * 2] = (v2f){v.z, v.w};
            }
        }
        // ---- stage B tile ----
        if (!TB) {
            // memory B[k][n]; vectorize over n
#pragma unroll
            for (int it = 0; it < (BK * BN / 4) / 256; ++it) {
                int t = tid + it * 256;
                int k = t / (BN / 4), nq = t % (BN / 4);
                v4f v = *(const v4f*)(B + (size_t)(k0 + k) * ldb + (n0 + 4 * nq));
                float* p = &Bs[(k >> 1) * BST + (4 * nq) * 2 + (k & 1)];
                p[0] = v.x; p[2] = v.y; p[4] = v.z; p[6] = v.w;
            }
        } else {
            // memory B[n][k]; vectorize over k
#pragma unroll
            for (int it = 0; it < (BN * BK / 4) / 256; ++it) {
                int t = tid + it * 256;
                int n = t / (BK / 4), q = t % (BK / 4);
                v4f v = *(const v4f*)(B + (size_t)(n0 + n) * ldb + (k0 + 4 * q));
                *(v2f*)&Bs[(2 * q) * BST + n * 2]     = (v2f){v.x, v.y};
                *(v2f*)&Bs[(2 * q + 1) * BST + n * 2] = (v2f){v.z, v.w};
            }
        }
        // ---- prefetch next k-tile into caches while this one computes ----
        if (k0 + BK < K) {
            if (TA) __builtin_prefetch(A + (size_t)(k0 + BK + (tid % BK)) * lda + m0 + (tid / BK) * 16, 0, 1);
            else    __builtin_prefetch(A + (size_t)(m0 + (tid % BM)) * lda + (k0 + BK), 0, 1);
            if (!TB) __builtin_prefetch(B + (size_t)(k0 + BK + (tid % BK)) * ldb + n0 + (tid / BK) * 16, 0, 1);
            else     __builtin_prefetch(B + (size_t)(n0 + (tid % BN)) * ldb + (k0 + BK), 0, 1);
        }
        __syncthreads();

#pragma unroll
        for (int kk = 0; kk < BK; kk += 4) {
            // A frag: lane holds M=l15, elems K = kk+2g, kk+2g+1  (one b64)
            v2f af[4];
#pragma unroll
            for (int tm = 0; tm < 4; ++tm) {
                int m = waveM * 64 + tm * 16 + l15;
                af[tm] = *(const v2f*)&As[(kk / 2 + g) * AST + m * 2];
            }
            v2f bf[2];
#pragma unroll
            for (int tn = 0; tn < 2; ++tn) {
                int n = waveN * 32 + tn * 16 + l15;
                bf[tn] = *(const v2f*)&Bs[(kk / 2 + g) * BST + n * 2];
            }
#pragma unroll
            for (int tm = 0; tm < 4; ++tm)
#pragma unroll
                for (int tn = 0; tn < 2; ++tn)
                    acc[tm][tn] = __builtin_amdgcn_wmma_f32_16x16x4_f32(
                        false, af[tm], false, bf[tn], (short)0, acc[tm][tn],
                        false, false);
        }
        __syncthreads();
    }

    // ---- epilogue: C/D elem r -> M = r + 8*g, N = l15 ----
#pragma unroll
    for (int tm = 0; tm < 4; ++tm) {
#pragma unroll
        for (int tn = 0; tn < 2; ++tn) {
            const int nb = n0 + waveN * 32 + tn * 16 + l15;
#pragma unroll
            for (int r = 0; r < 8; ++r) {
                const int mb = m0 + waveM * 64 + tm * 16 + r + 8 * g;
                const size_t off = (size_t)mb * ldc + nb;
                const float v = acc[tm][tn][r] * alpha;
                if (epi == EPI_STORE)            C[off] = v;
                else if (epi == EPI_IDENT)       C[off] = v + (mb == nb ? 1.0f : 0.0f);
                else if (epi == EPI_ACCUM)       C[off] = C[off] + v;
                else if (epi == EPI_BIAS)        C[off] = v + bias[nb];
                else /* EPI_BIAS_RELU */         C[off] = fmaxf(v + bias[nb], 0.0f);
            }
        }
    }
}

// ---------------------------------------------------------------------------
// In-place right-looking Cholesky (lower).  Zeroes the strict upper triangle
// so downstream GEMMs can treat the factor as dense.  Single block; the 17
// factorizations total ~6 GFLOP — negligible next to the GEMM chain.
// ---------------------------------------------------------------------------
__global__ __launch_bounds__(256) void chol_lower_kernel(float* __restrict__ A, int n)
{
    __shared__ float sdiag;
    const int tid = threadIdx.x;
    for (int k = 0; k < n; ++k) {
        if (tid == 0) {
            float dv = sqrtf(A[(size_t)k * n + k]);
            A[(size_t)k * n + k] = dv;
            sdiag = dv;
        }
        __syncthreads();
        const float dinv = 1.0f / sdiag;
        for (int i = k + 1 + tid; i < n; i += 256)
            A[(size_t)i * n + k] *= dinv;
        __syncthreads();
        for (int i = k + 1 + tid; i < n; i += 256) {
            const float lik = A[(size_t)i * n + k];
            float* rowi = A + (size_t)i * n;
            for (int j = k + 1; j <= i; ++j)
                rowi[j] -= lik * A[(size_t)j * n + k];
        }
        __syncthreads();
    }
    for (size_t idx = tid; idx < (size_t)n * n; idx += 256) {
        int i = (int)(idx / n), j = (int)(idx % n);
        if (j > i) A[idx] = 0.0f;
    }
}

// ---------------------------------------------------------------------------
// X = U^{-1} (upper), where U = Lᵀ and L is the lower factor stored in `Lm`.
// Column-parallel back-substitution; writes the full dense X (zeros below).
// ---------------------------------------------------------------------------
__global__ void triu_inv_lowT_kernel(const float* __restrict__ Lm,
                                     float* __restrict__ X, int n)
{
    const int j = blockIdx.x * blockDim.x + threadIdx.x;
    if (j >= n) return;
    for (int i = j + 1; i < n; ++i) X[(size_t)i * n + j] = 0.0f;
    X[(size_t)j * n + j] = 1.0f / Lm[(size_t)j * n + j];
    for (int i = j - 1; i >= 0; --i) {
        float s = 0.0f;
        for (int k = i + 1; k <= j; ++k)              // U[i][k] = L[k][i]
            s += Lm[(size_t)k * n + i] * X[(size_t)k * n + j];
        X[(size_t)i * n + j] = -s / Lm[(size_t)i * n + i];
    }
}

__global__ void set_identity_kernel(float* __restrict__ A, int n)
{
    size_t i = (size_t)blockIdx.x * blockDim.x + threadIdx.x;
    const size_t stride = (size_t)gridDim.x * blockDim.x;
    const size_t total = (size_t)n * n;
    for (; i < total; i += stride)
        A[i] = ((i / n) == (i % n)) ? 1.0f : 0.0f;
}

__global__ void copy_kernel(const float* __restrict__ src, float* __restrict__ dst,
                            size_t total)
{
    size_t i = (size_t)blockIdx.x * blockDim.x + threadIdx.x;
    const size_t stride = (size_t)gridDim.x * blockDim.x;
    for (; i < total; i += stride) dst[i] = src[i];
}

// ---------------------------------------------------------------------------
// Host-side sequencing (graph-capture safe: launches + pointer math only)
// ---------------------------------------------------------------------------
static inline void gemm(hipStream_t s, bool ta, bool tb,
                        const float* A, const float* B, const float* bias, float* C,
                        int M, int N, int K, int lda, int ldb, int ldc,
                        float alpha, int epi)
{
    dim3 grid(N / BN, M / BM), block(256);
    if (!ta && !tb)
        gemm_f32_wmma<false, false><<<grid, block, 0, s>>>(A, B, bias, C, K, lda, ldb, ldc, alpha, epi);
    else if (!ta && tb)
        gemm_f32_wmma<false, true><<<grid, block, 0, s>>>(A, B, bias, C, K, lda, ldb, ldc, alpha, epi);
    else if (ta && !tb)
        gemm_f32_wmma<true, false><<<grid, block, 0, s>>>(A, B, bias, C, K, lda, ldb, ldc, alpha, epi);
    else
        gemm_f32_wmma<true, true><<<grid, block, 0, s>>>(A, B, bias, C, K, lda, ldb, ldc, alpha, epi);
}

extern "C" void kernel_launch(void* const* d_in, const int* in_sizes, int n_in,
                              void* d_out, int out_size, void* d_ws, size_t ws_size,
                              hipStream_t stream)
{
    const float* x  = (const float*)d_in[0];
    const float* Va = (const float*)d_in[1];
    const float* ba = (const float*)d_in[2];
    const float* Vi = (const float*)d_in[3];
    const float* bi = (const float*)d_in[4];
    const float* Vb = (const float*)d_in[5];
    float* out = (float*)d_out;

    const int d     = in_sizes[2];            // 1024
    const int batch = in_sizes[0] / d;        // 4096
    const int nl    = in_sizes[4] / d;        // 8
    const size_t MAT = (size_t)d * d;

    // workspace carve-up (~64 MB of fp32)
    float* ws    = (float*)d_ws;
    float* d_inv = ws;  ws += MAT;   // omega_r^{-1} (upper, dense w/ zeros)
    float* a_w   = ws;  ws += MAT;   // a_weight
    float* gam   = ws;  ws += MAT;   // gamma
    float* sig   = ws;  ws += MAT;   // sigma_lower
    float* Tm    = ws;  ws += MAT;   // T (later: b_weight)
    float* Wm    = ws;  ws += MAT;   // W (later: Wb)
    float* chb   = ws;  ws += MAT;   // Cholesky scratch (R storage)
    float* tmp   = ws;  ws += MAT;   // sigma update scratch / left
    float* cur0  = ws;  ws += (size_t)batch * d;
    float* cur1  = ws;  ws += (size_t)batch * d;
    (void)ws_size; (void)n_in; (void)out_size;

    const float invLsq = 1.0f;       // 1 / L_SQ
    const float invCsq = 0.5f;       // 1 / C_SQ_INNER

    // --- layer A: R_a = chol(I + Vaᵀ Va / L²);  d_inv = R_a^{-1};  a_w = Va d_inv
    gemm(stream, true, false, Va, Va, nullptr, chb, d, d, d, d, d, d, invLsq, EPI_IDENT);
    chol_lower_kernel<<<1, 256, 0, stream>>>(chb, d);
    triu_inv_lowT_kernel<<<(d + 255) / 256, 256, 0, stream>>>(chb, d_inv, d);
    gemm(stream, false, false, Va, d_inv, nullptr, a_w, d, d, d, d, d, d, 1.0f, EPI_STORE);

    // --- first = x @ a_wᵀ + ba   (into d_out)
    gemm(stream, false, true, x, a_w, ba, out, batch, d, d, d, d, d, 1.0f, EPI_BIAS);

    // --- gamma = I
    set_identity_kernel<<<256, 256, 0, stream>>>(gam, d);

    const float* curIn = x;
    float* curA = cur0;
    float* curB = cur1;

    for (int i = 0; i < nl; ++i) {
        const float* V_i = Vi + (size_t)i * MAT;
        const float* b_i = bi + (size_t)i * d;

        // T = gamma @ d_inv        (d_inv == omega_r^{-1} here)
        gemm(stream, false, false, gam, d_inv, nullptr, Tm, d, d, d, d, d, d, 1.0f, EPI_STORE);
        // W = V_i @ omega_r^{-1}  (collapsed triangular solve)
        gemm(stream, false, false, V_i, d_inv, nullptr, Wm, d, d, d, d, d, d, 1.0f, EPI_STORE);
        // chb = I + W Wᵀ / c²
        gemm(stream, false, true, Wm, Wm, nullptr, chb, d, d, d, d, d, d, invCsq, EPI_IDENT);
        // cur = relu(cur @ Wᵀ + b_i)
        gemm(stream, false, true, curIn, Wm, b_i, curA, batch, d, d, d, d, d, 1.0f, EPI_BIAS_RELU);
        curIn = curA; { float* t = curA; curA = curB; curB = t; }
        // sigma_lower update
        if (i == 0) {
            copy_kernel<<<512, 256, 0, stream>>>(Tm, sig, MAT);
        } else {
            gemm(stream, false, true, sig, sig, nullptr, tmp, d, d, d, d, d, d, 1.0f, EPI_STORE);
            gemm(stream, false, true, Tm, Tm, nullptr, tmp, d, d, d, d, d, d, 1.0f, EPI_ACCUM);
            chol_lower_kernel<<<1, 256, 0, stream>>>(tmp, d);
            { float* t = sig; sig = tmp; tmp = t; }   // sigma <- chol(tmp)
        }
        // gamma = T @ Wᵀ
        gemm(stream, false, true, Tm, Wm, nullptr, gam, d, d, d, d, d, d, 1.0f, EPI_STORE);
        // omega_r = R = chol(chb);  d_inv = R^{-1}
        chol_lower_kernel<<<1, 256, 0, stream>>>(chb, d);
        triu_inv_lowT_kernel<<<(d + 255) / 256, 256, 0, stream>>>(chb, d_inv, d);
    }

    // --- tail: Wb = Vb @ omega_r^{-1}
    gemm(stream, false, false, Vb, d_inv, nullptr, Wm, d, d, d, d, d, d, 1.0f, EPI_STORE);
    // left = a_w @ sigma_lower
    gemm(stream, false, false, a_w, sig, nullptr, tmp, d, d, d, d, d, d, 1.0f, EPI_STORE);
    // R_sigma = chol(I + left leftᵀ);  Sinv = R_sigma^{-1}
    gemm(stream, false, true, tmp, tmp, nullptr, chb, d, d, d, d, d, d, 1.0f, EPI_IDENT);
    chol_lower_kernel<<<1, 256, 0, stream>>>(chb, d);
    triu_inv_lowT_kernel<<<(d + 255) / 256, 256, 0, stream>>>(chb, d_inv, d);
    // b_weight = Sinv @ Wb
    gemm(stream, false, false, d_inv, Wm, nullptr, Tm, d, d, d, d, d, d, 1.0f, EPI_STORE);
    // out += cur @ b_weightᵀ
    gemm(stream, false, true, curIn, Tm, nullptr, out, batch, d, d, d, d, d, 1.0f, EPI_ACCUM);
}